// Sinkhorn_30880814858386
// MI455X (gfx1250) — compile-verified
//
#include <hip/hip_runtime.h>

typedef __attribute__((ext_vector_type(2))) float v2f;
typedef __attribute__((ext_vector_type(8))) float v8f;

namespace {
constexpr int   MN    = 512;   // M == N == 512
constexpr int   JT    = 16;    // tiles per wave (16 rows x 256 cols slab)
constexpr int   RT    = 8;     // tiles resident in VGPRs
constexpr int   LT    = 8;     // tiles resident in LDS (JT - RT)
constexpr int   PAIRS = 10;    // 10 x (col step + row step) = 20 reference iters
constexpr float EPS   = 1e-10f;
}

// One workgroup (32 waves) per batch matrix. Wave w owns rows
// [16*(w>>1), +16) x columns [256*(w&1), +256) as 16 WMMA-layout 16x16 tiles:
// 8 in VGPRs, 8 in LDS (stored per-lane in B-operand pair layout so one
// ds_load_b64 yields a v2f WMMA operand; lanes only touch their own slots).
// Sinkhorn collapses to alternating masked matvecs (multiplicative scalings):
//   col step: c[n] = 1/sum_m a[m,n]*rr[m]   (WMMA f32 16x16x4 row-reduction)
//   row step: r[m] = 1/sum_n a[m,n]*cc[n]   (FMA + 16-lane butterfly)
// Final: out = (s0+eps)*r[m]*c[n] inside valid block, 0 outside (masks folded
// into rr/cc). Matrix stays entirely on-chip for all 20 iterations.
__global__ __launch_bounds__(1024) void sinkhorn_fused(
    const float* __restrict__ s,
    const int*   __restrict__ nrows,
    const int*   __restrict__ ncols,
    float*       __restrict__ out)
{
  // [tile][pair][wave][lane] : lane-contiguous v2f -> conflict-free ds_b64
  __shared__ v2f   ltile[LT][4][32][32];   // 256 KB
  __shared__ float partial[16][MN];        //  32 KB per-tile-row col partials
  __shared__ float vp[2][MN];              //   4 KB per-col-half row partials
  __shared__ float ccs[MN];                //   2 KB masked column scalings

  const int b     = blockIdx.x;
  const int tid   = threadIdx.x;
  const int w     = tid >> 5;          // wave id 0..31
  const int lane  = tid & 31;
  const int l16   = lane & 15;
  const int half8 = (lane >> 4) << 3;  // 0 for lanes 0-15, 8 for lanes 16-31
  const int trow  = w >> 1;            // tile row 0..15
  const int chalf = w & 1;             // column half 0..1
  const int rbase = 16 * trow;
  const int cbase = 256 * chalf;

  const int nr = nrows[b];
  const int nc = ncols[b];

  const float* __restrict__ sb = s   + (size_t)b * MN * MN;
  float*       __restrict__ ob = out + (size_t)b * MN * MN;

  // ---- load resident tiles; pair (g2) = rows {2g2, 2g2+1} (+half8), col l16
  v2f A[RT][4];
#pragma unroll
  for (int j = 0; j < JT; ++j) {
    const int col = cbase + 16 * j + l16;
#pragma unroll
    for (int g2 = 0; g2 < 4; ++g2) {
      const int r0 = rbase + 2 * g2 + half8;
      v2f p;
      p.x = __builtin_nontemporal_load(&sb[(size_t)r0 * MN + col]) + EPS;
      p.y = __builtin_nontemporal_load(&sb[(size_t)(r0 + 1) * MN + col]) + EPS;
      if (j < RT) A[j][g2] = p;
      else        ltile[j - RT][g2][w][lane] = p;   // private slot, no barrier
    }
  }

  // rrp[g2]: masked row scalings, broadcast-uniform per 16-lane half; the
  // register pair IS the WMMA A(16x4) operand (VGPR0=K0/K2, VGPR1=K1/K3).
  v2f rrp[4];
#pragma unroll
  for (int g2 = 0; g2 < 4; ++g2) {
    rrp[g2].x = (rbase + 2 * g2 + half8     < nr) ? 1.0f : 0.0f;
    rrp[g2].y = (rbase + 2 * g2 + 1 + half8 < nr) ? 1.0f : 0.0f;
  }

  for (int it = 0; it < PAIRS; ++it) {
    // ======== column step: u[n] = sum_m a[m,n]*rr[m]  (WMMA row-reduce)
#pragma unroll
    for (int j = 0; j < JT; ++j) {
      v8f d = {0.f, 0.f, 0.f, 0.f, 0.f, 0.f, 0.f, 0.f};
#pragma unroll
      for (int g2 = 0; g2 < 4; ++g2) {
        const v2f bv = (j < RT) ? A[j][g2] : ltile[j - RT][g2][w][lane];
        d = __builtin_amdgcn_wmma_f32_16x16x4_f32(
                false, rrp[g2], false, bv, (short)0, d, false, false);
      }
      if (lane < 16) partial[trow][cbase + 16 * j + l16] = d[0];
    }
    __syncthreads();

    if (tid < MN) {
      float u = 0.f;
#pragma unroll
      for (int k = 0; k < 16; ++k) u += partial[k][tid];
      ccs[tid] = (tid < nc) ? (1.0f / u) : 0.0f;
    }
    __syncthreads();

    // ======== row step: v[m] = sum_n a[m,n]*cc[n]  (FMA + lane butterfly)
    float vacc[8] = {0.f, 0.f, 0.f, 0.f, 0.f, 0.f, 0.f, 0.f};
#pragma unroll
    for (int j = 0; j < JT; ++j) {
      const float cv = ccs[cbase + 16 * j + l16];
#pragma unroll
      for (int g2 = 0; g2 < 4; ++g2) {
        const v2f p = (j < RT) ? A[j][g2] : ltile[j - RT][g2][w][lane];
        vacc[2 * g2]     = fmaf(p.x, cv, vacc[2 * g2]);
        vacc[2 * g2 + 1] = fmaf(p.y, cv, vacc[2 * g2 + 1]);
      }
    }
#pragma unroll
    for (int g = 0; g < 8; ++g) {
#pragma unroll
      for (int m = 1; m <= 8; m <<= 1)
        vacc[g] += __shfl_xor(vacc[g], m, 32);  // reduce within 16-lane half
    }
#pragma unroll
    for (int g = 0; g < 8; ++g)
      if (l16 == g) vp[chalf][rbase + g + half8] = vacc[g];
    __syncthreads();

    // combine column-halves; r[m] = (m<nr) ? 1/v[m] : 0 (uniform per half)
#pragma unroll
    for (int g2 = 0; g2 < 4; ++g2) {
      const int r0 = rbase + 2 * g2 + half8;
      const float v0 = vp[0][r0] + vp[1][r0];
      const float v1 = vp[0][r0 + 1] + vp[1][r0 + 1];
      rrp[g2].x = (r0     < nr) ? (1.0f / v0) : 0.0f;
      rrp[g2].y = (r0 + 1 < nr) ? (1.0f / v1) : 0.0f;
    }
    // hazards: partial re-written next col step (fenced by sync after u-read);
    // ccs re-written next u-compute (fenced by sync after next col step);
    // vp re-written next row step (fenced by both syncs above).
  }

  // ======== final: out = (s0+eps)*r[m]*c[n]; masks give 0 outside block
#pragma unroll
  for (int j = 0; j < JT; ++j) {
    const int col = cbase + 16 * j + l16;
    const float cv = ccs[col];
#pragma unroll
    for (int g2 = 0; g2 < 4; ++g2) {
      const int r0 = rbase + 2 * g2 + half8;
      const v2f p = (j < RT) ? A[j][g2] : ltile[j - RT][g2][w][lane];
      __builtin_nontemporal_store(p.x * rrp[g2].x * cv,
                                  &ob[(size_t)r0 * MN + col]);
      __builtin_nontemporal_store(p.y * rrp[g2].y * cv,
                                  &ob[(size_t)(r0 + 1) * MN + col]);
    }
  }
}

extern "C" void kernel_launch(void* const* d_in, const int* in_sizes, int n_in,
                              void* d_out, int out_size, void* d_ws, size_t ws_size,
                              hipStream_t stream) {
  (void)n_in; (void)out_size; (void)d_ws; (void)ws_size;
  const float* s     = (const float*)d_in[0];
  const int*   nrows = (const int*)d_in[1];
  const int*   ncols = (const int*)d_in[2];
  float*       out   = (float*)d_out;
  const int B = in_sizes[1];  // one nrows entry per batch matrix
  sinkhorn_fused<<<dim3(B), dim3(1024), 0, stream>>>(s, nrows, ncols, out);
}